// MultiBoxLoss_57028575756872
// MI455X (gfx1250) — compile-verified
//
#include <hip/hip_runtime.h>
#include <hip/hip_bf16.h>
#include <math.h>

#define NUM_CLASSES 80
#define BATCH 8
#define MAX_BOXES 32
#define NA 76725
#define NBLK ((NA + 255) / 256)      /* 300 blocks of 256 anchors per image */
#define TOTP (BATCH * NBLK)          /* 2400 partials per quantity */

typedef __attribute__((ext_vector_type(2))) float v2f;
typedef __attribute__((ext_vector_type(8))) float v8f;

// Fused kernel: per-anchor IoU assignment + focal classification loss +
// smooth-L1 regression loss. Per-block partial sums (cls, reg, pos_count)
// written to workspace; wave-level reduction done with V_WMMA_F32_16X16X4_F32.
__global__ __launch_bounds__(256) void multibox_main(
    const float* __restrict__ cls_pred,   // [B, 80, NA]
    const float* __restrict__ reg_pred,   // [B, 4, NA]
    const float* __restrict__ annots,     // [B, 32, 5]
    const float* __restrict__ anchors,    // [NA, 4]
    float* __restrict__ partial)          // [3 * TOTP]
{
    __shared__ float sbox[MAX_BOXES * 5];
    __shared__ float sWCls[8];
    __shared__ float sWReg[8];
    __shared__ int   sPos;

    const int tid = threadIdx.x;
    const int b   = blockIdx.y;
    const int a   = blockIdx.x * 256 + tid;

    if (tid < MAX_BOXES * 5) sbox[tid] = annots[b * MAX_BOXES * 5 + tid];
    if (tid == 0) sPos = 0;
    __syncthreads();

    float clsAcc = 0.0f;
    float regAcc = 0.0f;
    bool  positive = false;

    if (a < NA) {
        const float a0 = anchors[a * 4 + 0];
        const float a1 = anchors[a * 4 + 1];
        const float a2 = anchors[a * 4 + 2];
        const float a3 = anchors[a * 4 + 3];
        const float aw  = a2 - a0;
        const float ah  = a3 - a1;
        const float acx = a0 + 0.5f * aw;
        const float acy = a1 + 0.5f * ah;
        const float aarea = aw * ah;

        // ---- IoU max / argmax over 32 boxes (first-max tie-break via strict >)
        float best = -2.0f;
        int   bj   = 0;
        #pragma unroll 4
        for (int j = 0; j < MAX_BOXES; ++j) {
            const float bx0 = sbox[j * 5 + 0];
            const float bx1 = sbox[j * 5 + 1];
            const float bx2 = sbox[j * 5 + 2];
            const float bx3 = sbox[j * 5 + 3];
            const float bc  = sbox[j * 5 + 4];
            float v;
            if (bc != -1.0f) {
                const float barea = (bx2 - bx0) * (bx3 - bx1);
                float iw = fminf(a2, bx2) - fmaxf(a0, bx0); iw = fmaxf(iw, 0.0f);
                float ih = fminf(a3, bx3) - fmaxf(a1, bx1); ih = fmaxf(ih, 0.0f);
                const float inter = iw * ih;
                const float uni   = fmaxf(aarea + barea - inter, 1e-7f);
                v = inter / uni;
            } else {
                v = -1.0f;
            }
            if (v > best) { best = v; bj = j; }
        }
        positive = (best >= 0.5f);
        const int tc = positive ? (int)sbox[bj * 5 + 4] : -1;

        // ---- focal classification loss over 80 classes (coalesced stride-NA reads)
        const float* cp = cls_pred + (size_t)b * NUM_CLASSES * NA + a;
        for (int c = 0; c < NUM_CLASSES; ++c) {
            if (c + 4 < NUM_CLASSES)
                __builtin_prefetch(cp + (size_t)(c + 4) * NA, 0, 1);
            float p = cp[(size_t)c * NA];
            p = fminf(fmaxf(p, 1e-7f), 1.0f - 1e-7f);
            float t;
            if (c == tc) {
                const float q = 1.0f - p;
                t = 0.25f * q * q * (-logf(p));          // target = 1
            } else {
                t = 0.75f * p * p * (-log1pf(-p));       // target = 0
            }
            clsAcc += t;
        }

        // ---- smooth-L1 regression loss (positives only; mask-equivalent)
        if (positive) {
            const float g0 = sbox[bj * 5 + 0];
            const float g1 = sbox[bj * 5 + 1];
            const float g2 = sbox[bj * 5 + 2];
            const float g3 = sbox[bj * 5 + 3];
            const float gwr = g2 - g0;
            const float ghr = g3 - g1;
            const float gcx = g0 + 0.5f * gwr;
            const float gcy = g1 + 0.5f * ghr;
            const float gw  = fmaxf(gwr, 1.0f);
            const float gh  = fmaxf(ghr, 1.0f);
            float t4[4];
            t4[0] = ((gcx - acx) / aw) / 0.1f;
            t4[1] = ((gcy - acy) / ah) / 0.1f;
            t4[2] = logf(gw / aw) / 0.2f;
            t4[3] = logf(gh / ah) / 0.2f;
            const float* rp = reg_pred + (size_t)b * 4 * NA + a;
            #pragma unroll
            for (int k = 0; k < 4; ++k) {
                const float diff = fabsf(t4[k] - rp[(size_t)k * NA]);
                regAcc += (diff <= (1.0f / 9.0f)) ? (4.5f * diff * diff)
                                                  : (diff - (0.5f / 9.0f));
            }
        }
    }

    if (positive) atomicAdd(&sPos, 1);

    // ---- wave-level reduction with V_WMMA_F32_16X16X4_F32.
    // Pack the wave's 32 cls partials into A rows 0-7 and the 32 reg partials
    // into A rows 8-15 (documented 16x4 f32 A layout). B = all-ones, so
    // D[m][n] = rowsum(A[m]) regardless of B's lane layout. By the C/D layout,
    // lanes 0-15 then hold rowsums of M=0..7 (cls) in their 8 VGPRs and
    // lanes 16-31 hold rowsums of M=8..15 (reg).
    const int lane = tid & 31;
    const int wv   = tid >> 5;
    const int s0   = 2 * (lane & 7) + (lane & 16);
    const float c0 = __shfl(clsAcc, s0,     32);
    const float c1 = __shfl(clsAcc, s0 + 1, 32);
    const float r0 = __shfl(regAcc, s0,     32);
    const float r1 = __shfl(regAcc, s0 + 1, 32);
    const bool takeCls = (lane & 15) < 8;

    v2f av; av[0] = takeCls ? c0 : r0; av[1] = takeCls ? c1 : r1;
    v2f bv; bv[0] = 1.0f;              bv[1] = 1.0f;
    v8f cv = {};
    v8f dv = __builtin_amdgcn_wmma_f32_16x16x4_f32(
        /*neg_a=*/false, av, /*neg_b=*/false, bv,
        /*c_mod=*/(short)0, cv, /*reuse_a=*/false, /*reuse_b=*/false);

    float s = dv[0] + dv[1] + dv[2] + dv[3] + dv[4] + dv[5] + dv[6] + dv[7];
    const float waveCls = __shfl(s, 0,  32);
    const float waveReg = __shfl(s, 16, 32);
    if (lane == 0) { sWCls[wv] = waveCls; sWReg[wv] = waveReg; }
    __syncthreads();

    if (tid == 0) {
        float cs = 0.0f, rs = 0.0f;
        #pragma unroll
        for (int w = 0; w < 8; ++w) { cs += sWCls[w]; rs += sWReg[w]; }
        const int idx = b * NBLK + blockIdx.x;
        partial[idx]            = cs;
        partial[TOTP + idx]     = rs;
        partial[2 * TOTP + idx] = (float)sPos;
    }
}

// Deterministic fixed-order finalization: per-image sums (double), apply
// num_pos normalization, batch mean -> scalar.
__global__ __launch_bounds__(64) void multibox_final(
    const float* __restrict__ partial, float* __restrict__ out)
{
    __shared__ double sc[BATCH], sr[BATCH], sp[BATCH];
    const int t = threadIdx.x;
    if (t < BATCH) {
        double c = 0.0, r = 0.0, p = 0.0;
        for (int i = 0; i < NBLK; ++i) {
            c += (double)partial[t * NBLK + i];
            r += (double)partial[TOTP + t * NBLK + i];
            p += (double)partial[2 * TOTP + t * NBLK + i];
        }
        sc[t] = c; sr[t] = r; sp[t] = p;
    }
    __syncthreads();
    if (t == 0) {
        double acc = 0.0;
        for (int bb = 0; bb < BATCH; ++bb) {
            const double np   = sp[bb];
            const double clsL = sc[bb] / fmax(np, 1.0);
            const double regL = (np > 0.0) ? (sr[bb] / fmax(4.0 * np, 1.0)) : 0.0;
            acc += clsL + regL;
        }
        out[0] = (float)(acc / (double)BATCH);
    }
}

extern "C" void kernel_launch(void* const* d_in, const int* in_sizes, int n_in,
                              void* d_out, int out_size, void* d_ws, size_t ws_size,
                              hipStream_t stream) {
    (void)in_sizes; (void)n_in; (void)out_size; (void)ws_size;
    const float* cls_pred = (const float*)d_in[0];
    const float* reg_pred = (const float*)d_in[1];
    const float* annots   = (const float*)d_in[2];
    const float* anchors  = (const float*)d_in[3];
    float* partial = (float*)d_ws;   // needs 3 * 2400 * 4 = 28.8 KB
    float* outp    = (float*)d_out;

    dim3 grid(NBLK, BATCH);
    hipLaunchKernelGGL(multibox_main, grid, dim3(256), 0, stream,
                       cls_pred, reg_pred, annots, anchors, partial);
    hipLaunchKernelGGL(multibox_final, dim3(1), dim3(64), 0, stream,
                       partial, outp);
}